// GAT_69544110457527
// MI455X (gfx1250) — compile-verified
//
#include <hip/hip_runtime.h>
#include <math.h>

// ---------------------------------------------------------------------------
// Problem constants (from reference)
// ---------------------------------------------------------------------------
#define N_NODES 50000
#define N_EDGES 800000
#define E_TOT   (N_EDGES + N_NODES)   // with self loops = 850000
#define IN_DIM  128
#define FEAT    64                    // HEADS*HID == OUT_DIM == 64 for both layers
#define NEG_SLOPE 0.2f

typedef float v2f __attribute__((ext_vector_type(2)));
typedef float v8f __attribute__((ext_vector_type(8)));

// ---------------------------------------------------------------------------
// fill a float buffer with a constant (used for -inf max-init and 0 accum)
// ---------------------------------------------------------------------------
__global__ void gat_fill_k(float* __restrict__ p, float v, int n) {
    int i = blockIdx.x * blockDim.x + threadIdx.x;
    if (i < n) p[i] = v;
}

// ---------------------------------------------------------------------------
// C[M x N] = A[M x K] @ B[K x N], full fp32 via V_WMMA_F32_16X16X4_F32.
// One wave (32 threads) per 16x16 output tile. M % 16 == 0, N % 16 == 0,
// K % 4 == 0 (holds: M=50000, N=64, K=128/64). EXEC is all ones (no
// divergence before/around the WMMA), as the ISA requires.
//
// ISA VGPR layouts (CDNA5 ISA):
//   A 16x4 : lanes 0-15 -> M=lane,  v0=K0,v1=K1 ; lanes 16-31 -> v0=K2,v1=K3
//   B 4x16 : lanes 0-15 -> N=lane,  v0=K0,v1=K1 ; lanes 16-31 -> v0=K2,v1=K3
//   C/D    : vgpr r -> M = r + 8*(lane/16), N = lane%16
// ---------------------------------------------------------------------------
__global__ __launch_bounds__(32) void gat_gemm_wmma_f32_k(
    const float* __restrict__ A, const float* __restrict__ B,
    float* __restrict__ C, int K, int N) {
    const int lane = threadIdx.x & 31;
    const int half = lane >> 4;     // 0: K pair {0,1}, 1: K pair {2,3}
    const int ln   = lane & 15;     // M index for A, N index for B/C/D
    const int m0   = blockIdx.x << 4;
    const int n0   = blockIdx.y << 4;

    const float* __restrict__ arow = A + (size_t)(m0 + ln) * K;

    v8f acc = {};
    for (int k0 = 0; k0 < K; k0 += 4) {
        const int ka = k0 + 2 * half;
        v2f a, b;
        a.x = arow[ka];
        a.y = arow[ka + 1];
        b.x = B[(size_t)ka * N + n0 + ln];
        b.y = B[(size_t)(ka + 1) * N + n0 + ln];
        acc = __builtin_amdgcn_wmma_f32_16x16x4_f32(
            /*neg_a=*/false, a, /*neg_b=*/false, b,
            /*c_mod=*/(short)0, acc, /*reuse_a=*/false, /*reuse_b=*/false);
    }

    float* __restrict__ crow = C + (size_t)(m0 + half * 8) * N + n0 + ln;
#pragma unroll
    for (int r = 0; r < 8; ++r) crow[(size_t)r * N] = acc[r];
}

// ---------------------------------------------------------------------------
// Per-(node, head) attention score dots:
//   s_src[n,h] = sum_c h[n, h*ch + c] * a_src[h, c]   (same for dst)
// ---------------------------------------------------------------------------
__global__ void gat_score_k(const float* __restrict__ h,
                            const float* __restrict__ asrc,
                            const float* __restrict__ adst,
                            float* __restrict__ ssrc, float* __restrict__ sdst,
                            int total, int heads, int ch) {
    int i = blockIdx.x * blockDim.x + threadIdx.x;
    if (i >= total) return;
    int n  = i / heads;
    int hh = i - n * heads;
    const float* row = h + (size_t)n * heads * ch + hh * ch;
    const float* as  = asrc + hh * ch;
    const float* ad  = adst + hh * ch;
    float ss = 0.f, sd = 0.f;
    for (int c = 0; c < ch; ++c) {
        float v = row[c];
        ss += v * as[c];
        sd += v * ad[c];
    }
    ssrc[i] = ss;
    sdst[i] = sd;
}

// Self-loop aware edge decode: edges [0,E) from edge_index, [E, E+N) are loops.
__device__ __forceinline__ void gat_edge_sd(const int* __restrict__ ei, int E,
                                            int e, int& s, int& d) {
    if (e < E) { s = ei[e]; d = ei[E + e]; }
    else       { s = e - E; d = s; }
}

__device__ __forceinline__ float gat_leaky(float v) {
    return v > 0.f ? v : NEG_SLOPE * v;
}

// ---------------------------------------------------------------------------
// Pass 1: segment max over destination (float atomic max -> GLOBAL_ATOMIC_MAX_NUM_F32)
// ---------------------------------------------------------------------------
__global__ void gat_edge_max_k(const int* __restrict__ ei, int E, int etot,
                               const float* __restrict__ ssrc,
                               const float* __restrict__ sdst,
                               float* __restrict__ mmax, int heads) {
    int i = blockIdx.x * blockDim.x + threadIdx.x;
    if (i >= etot * heads) return;
    int e  = i / heads;
    int hh = i - e * heads;
    int s, d;
    gat_edge_sd(ei, E, e, s, d);
    float v = gat_leaky(ssrc[s * heads + hh] + sdst[d * heads + hh]);
    atomicMax(&mmax[d * heads + hh], v);
}

// ---------------------------------------------------------------------------
// Pass 2: segment sum of exp(e - m[dst])
// ---------------------------------------------------------------------------
__global__ void gat_edge_sum_k(const int* __restrict__ ei, int E, int etot,
                               const float* __restrict__ ssrc,
                               const float* __restrict__ sdst,
                               const float* __restrict__ mmax,
                               float* __restrict__ den, int heads) {
    int i = blockIdx.x * blockDim.x + threadIdx.x;
    if (i >= etot * heads) return;
    int e  = i / heads;
    int hh = i - e * heads;
    int s, d;
    gat_edge_sd(ei, E, e, s, d);
    float v = gat_leaky(ssrc[s * heads + hh] + sdst[d * heads + hh]);
    float p = expf(v - mmax[d * heads + hh]);
    atomicAdd(&den[d * heads + hh], p);
}

// ---------------------------------------------------------------------------
// Pass 3: weighted scatter-add. 64 consecutive lanes handle one edge's 64
// features -> 256B coalesced gather + 256B coalesced atomic row update.
// ---------------------------------------------------------------------------
__global__ void gat_edge_scatter_k(const int* __restrict__ ei, int E, int etot,
                                   const float* __restrict__ ssrc,
                                   const float* __restrict__ sdst,
                                   const float* __restrict__ mmax,
                                   const float* __restrict__ den,
                                   const float* __restrict__ hfeat,
                                   float* __restrict__ outacc,
                                   int heads, int ch) {
    int i = blockIdx.x * blockDim.x + threadIdx.x;   // e*64 + j  (max 54.4M)
    if (i >= etot * FEAT) return;
    int e = i >> 6;
    int j = i & (FEAT - 1);
    int s, d;
    gat_edge_sd(ei, E, e, s, d);
    int hh = j / ch;                                  // ch=8 -> j>>3 ; ch=64 -> 0
    float v     = gat_leaky(ssrc[s * heads + hh] + sdst[d * heads + hh]);
    float p     = expf(v - mmax[d * heads + hh]);
    float alpha = p / (den[d * heads + hh] + 1e-16f);
    atomicAdd(&outacc[d * FEAT + j], hfeat[s * FEAT + j] * alpha);
}

// ---------------------------------------------------------------------------
// out[i] = in[i] + bias[i % 64]; optional ELU
// ---------------------------------------------------------------------------
__global__ void gat_bias_act_k(const float* __restrict__ in,
                               const float* __restrict__ bias,
                               float* __restrict__ out, int n, int do_elu) {
    int i = blockIdx.x * blockDim.x + threadIdx.x;
    if (i >= n) return;
    float t = in[i] + bias[i & (FEAT - 1)];
    if (do_elu) t = t > 0.f ? t : expm1f(t);
    out[i] = t;
}

// ---------------------------------------------------------------------------
// Host-side launcher
// ---------------------------------------------------------------------------
static inline int gat_cdiv(int a, int b) { return (a + b - 1) / b; }

extern "C" void kernel_launch(void* const* d_in, const int* in_sizes, int n_in,
                              void* d_out, int out_size, void* d_ws, size_t ws_size,
                              hipStream_t stream) {
    (void)in_sizes; (void)n_in; (void)out_size; (void)ws_size;

    const float* x   = (const float*)d_in[0];
    const int*   ei  = (const int*)  d_in[1];   // [2, E] row-major: src then dst
    const float* W1  = (const float*)d_in[2];
    const float* as1 = (const float*)d_in[3];
    const float* ad1 = (const float*)d_in[4];
    const float* b1  = (const float*)d_in[5];
    const float* W2  = (const float*)d_in[6];
    const float* as2 = (const float*)d_in[7];
    const float* ad2 = (const float*)d_in[8];
    const float* b2  = (const float*)d_in[9];
    float* out = (float*)d_out;

    const int N = N_NODES, E = N_EDGES, ET = E_TOT;
    const float NEG_INF = -__builtin_huge_valf();

    // carve scratch from d_ws (~58 MB total)
    float* ws   = (float*)d_ws;
    float* h1   = ws; ws += N * FEAT;   // x @ W1
    float* ss1  = ws; ws += N * 8;
    float* sd1  = ws; ws += N * 8;
    float* m1   = ws; ws += N * 8;
    float* dn1  = ws; ws += N * 8;
    float* agg1 = ws; ws += N * FEAT;   // layer-1 aggregation -> +b1 -> ELU (in place)
    float* h2   = ws; ws += N * FEAT;   // agg1 @ W2
    float* ss2  = ws; ws += N;
    float* sd2  = ws; ws += N;
    float* m2   = ws; ws += N;
    float* dn2  = ws; ws += N;
    float* agg2 = ws; ws += N * FEAT;   // layer-2 aggregation

    const int B = 256;

    // init accumulators every call (harness does not re-poison between replays)
    gat_fill_k<<<gat_cdiv(N * 8, B), B, 0, stream>>>(m1, NEG_INF, N * 8);
    gat_fill_k<<<gat_cdiv(N * 8, B), B, 0, stream>>>(dn1, 0.f, N * 8);
    gat_fill_k<<<gat_cdiv(N * FEAT, B), B, 0, stream>>>(agg1, 0.f, N * FEAT);
    gat_fill_k<<<gat_cdiv(N, B), B, 0, stream>>>(m2, NEG_INF, N);
    gat_fill_k<<<gat_cdiv(N, B), B, 0, stream>>>(dn2, 0.f, N);
    gat_fill_k<<<gat_cdiv(N * FEAT, B), B, 0, stream>>>(agg2, 0.f, N * FEAT);

    // ---------------- layer 1 (heads=8, ch=8) ----------------
    gat_gemm_wmma_f32_k<<<dim3(N / 16, FEAT / 16), 32, 0, stream>>>(x, W1, h1, IN_DIM, FEAT);
    gat_score_k<<<gat_cdiv(N * 8, B), B, 0, stream>>>(h1, as1, ad1, ss1, sd1, N * 8, 8, 8);
    gat_edge_max_k<<<gat_cdiv(ET * 8, B), B, 0, stream>>>(ei, E, ET, ss1, sd1, m1, 8);
    gat_edge_sum_k<<<gat_cdiv(ET * 8, B), B, 0, stream>>>(ei, E, ET, ss1, sd1, m1, dn1, 8);
    gat_edge_scatter_k<<<gat_cdiv(ET * FEAT, B), B, 0, stream>>>(
        ei, E, ET, ss1, sd1, m1, dn1, h1, agg1, 8, 8);
    gat_bias_act_k<<<gat_cdiv(N * FEAT, B), B, 0, stream>>>(agg1, b1, agg1, N * FEAT, 1);

    // ---------------- layer 2 (heads=1, ch=64) ----------------
    gat_gemm_wmma_f32_k<<<dim3(N / 16, FEAT / 16), 32, 0, stream>>>(agg1, W2, h2, FEAT, FEAT);
    gat_score_k<<<gat_cdiv(N, B), B, 0, stream>>>(h2, as2, ad2, ss2, sd2, N, 1, FEAT);
    gat_edge_max_k<<<gat_cdiv(ET, B), B, 0, stream>>>(ei, E, ET, ss2, sd2, m2, 1);
    gat_edge_sum_k<<<gat_cdiv(ET, B), B, 0, stream>>>(ei, E, ET, ss2, sd2, m2, dn2, 1);
    gat_edge_scatter_k<<<gat_cdiv(ET * FEAT, B), B, 0, stream>>>(
        ei, E, ET, ss2, sd2, m2, dn2, h2, agg2, 1, FEAT);
    gat_bias_act_k<<<gat_cdiv(N * FEAT, B), B, 0, stream>>>(agg2, b2, out, N * FEAT, 0);
}